// PatchPooling_62448824484364
// MI455X (gfx1250) — compile-verified
//
#include <hip/hip_runtime.h>
#include <hip/hip_bf16.h>

// Problem constants (match reference)
#define BB 8
#define SS 8192
#define HH 512
#define DD 768
#define PP 128
#define CS 256          // sequence chunk per pooling block
#define PT 128          // pooling threads per block (4 channels each)

typedef __attribute__((ext_vector_type(2))) float v2f;
typedef __attribute__((ext_vector_type(4))) float v4f;
typedef __attribute__((ext_vector_type(8))) float v8f;

// ---------------------------------------------------------------------------
// Kernel 0: zero the workspace accumulators (sums [B*P,H] + counts [B*P]).
// ---------------------------------------------------------------------------
__global__ void pp_zero_ws(float* __restrict__ ws, int n) {
    int i = blockIdx.x * blockDim.x + threadIdx.x;
    if (i < n) ws[i] = 0.0f;
}

// ---------------------------------------------------------------------------
// Kernel 1: segment (patch) sums. patch_ids is sorted per batch, so we keep a
// running accumulator per thread (4 channels each, B128 loads) and only
// atomicAdd to the global sums buffer at patch boundaries (avg run = S/P = 64).
// Patch ids for the chunk are staged through LDS once.
// ---------------------------------------------------------------------------
__global__ void pp_pool(const float* __restrict__ x,       // [B,S,H]
                        const int*   __restrict__ pid,     // [B,S]
                        float* __restrict__ sums,          // [B*P,H]
                        float* __restrict__ counts) {      // [B*P]
    __shared__ int spid[CS];
    const int chunks_per_b = SS / CS;
    const int b     = blockIdx.x / chunks_per_b;
    const int chunk = blockIdx.x % chunks_per_b;
    const int s0    = chunk * CS;
    const int tid   = threadIdx.x;

    spid[tid]      = pid[b * SS + s0 + tid];
    spid[tid + PT] = pid[b * SS + s0 + tid + PT];
    __syncthreads();

    const int h0 = tid * 4;                       // 128 threads x 4 ch = 512
    const float* base = x + ((size_t)b * SS + s0) * HH + h0;

    v4f acc = {};
    int cur = spid[0];
    int run = 0;
    for (int s = 0; s < CS; ++s) {
        const int p = spid[s];                    // uniform across block
        if (p != cur) {                           // uniform branch
            float* dst = sums + (size_t)(b * PP + cur) * HH + h0;
            atomicAdd(dst + 0, acc.x);
            atomicAdd(dst + 1, acc.y);
            atomicAdd(dst + 2, acc.z);
            atomicAdd(dst + 3, acc.w);
            if (tid == 0) atomicAdd(&counts[b * PP + cur], (float)run);
            acc = {}; run = 0; cur = p;
        }
        v4f v = *(const v4f*)(base + (size_t)s * HH);   // global_load_b128
        acc.x += v.x; acc.y += v.y; acc.z += v.z; acc.w += v.w;
        ++run;
    }
    float* dst = sums + (size_t)(b * PP + cur) * HH + h0;
    atomicAdd(dst + 0, acc.x);
    atomicAdd(dst + 1, acc.y);
    atomicAdd(dst + 2, acc.z);
    atomicAdd(dst + 3, acc.w);
    if (tid == 0) atomicAdd(&counts[b * PP + cur], (float)run);
}

// ---------------------------------------------------------------------------
// Kernel 2: (sums @ W) * (1/count) + bias using V_WMMA_F32_16X16X4_F32.
// M = B*P = 1024, K = H = 512, N = D = 768.
// One 16x16 output tile per wave; 8 waves per 256-thread block.
// The divide-by-count is applied per OUTPUT ROW in the epilogue, so the
// K-loop is pure loads + WMMA on the natively-pipelined D->C accumulate path.
// ---------------------------------------------------------------------------
__global__ void pp_gemm_wmma(const float* __restrict__ sums,    // [1024,512]
                             const float* __restrict__ counts,  // [1024]
                             const float* __restrict__ Wp,      // [512,768]
                             const float* __restrict__ bias,    // [768]
                             float* __restrict__ out) {         // [1024,768]
    const int lane = threadIdx.x & 31;
    const int wave = threadIdx.x >> 5;
    const int tile = blockIdx.x * 8 + wave;       // 3072 tiles total
    const int tm = tile % 64;                     // 64 M-tiles
    const int tn = tile / 64;                     // 48 N-tiles
    const int m0 = tm * 16;
    const int n0 = tn * 16;

    const int l15 = lane & 15;
    const int hi  = lane >> 4;                    // 0: lanes 0-15, 1: 16-31
    const int mA  = m0 + l15;                     // A row owned by this lane
    const int kA  = hi * 2;                       // K sub-offset (0 or 2)
    const int n   = n0 + l15;                     // output column of this lane

    const float* arow = sums + (size_t)mA * HH + kA;
    const float* bcol = Wp   + (size_t)kA * DD + n;

    v8f acc = {};
    #pragma unroll 8
    for (int kk = 0; kk < HH; kk += 4) {
        v2f a = *(const v2f*)(arow + kk);         // A[mA][kk+kA .. kk+kA+1]
        v2f bf;
        bf.x = bcol[(size_t)kk * DD];             // B[kk+kA  ][n]
        bf.y = bcol[(size_t)kk * DD + DD];        // B[kk+kA+1][n]
        // 8 args: (neg_a, A, neg_b, B, c_mod, C, reuse_a, reuse_b)
        acc = __builtin_amdgcn_wmma_f32_16x16x4_f32(
            false, a, false, bf, (short)0, acc, false, false);
    }

    // Epilogue: per-row mean scale + bias, then store.
    const float bb = bias[n];
    const int   row0 = m0 + hi * 8;               // first of this lane's 8 rows
    float* orow = out + (size_t)row0 * DD + n;
    #pragma unroll
    for (int r = 0; r < 8; ++r) {
        const float scale = 1.0f / fmaxf(counts[row0 + r], 1.0f);
        orow[(size_t)r * DD] = fmaf(acc[r], scale, bb);
    }
}

// ---------------------------------------------------------------------------
extern "C" void kernel_launch(void* const* d_in, const int* in_sizes, int n_in,
                              void* d_out, int out_size, void* d_ws, size_t ws_size,
                              hipStream_t stream) {
    const float* byte_hiddens = (const float*)d_in[0];   // [8,8192,512] f32
    const float* W_proj       = (const float*)d_in[1];   // [512,768]    f32
    const float* b_proj       = (const float*)d_in[2];   // [768]        f32
    const int*   patch_ids    = (const int*)  d_in[3];   // [8,8192]     i32
    float*       out          = (float*)d_out;           // [8,128,768]  f32

    float* sums   = (float*)d_ws;                        // [1024,512] = 2 MiB
    float* counts = sums + (size_t)BB * PP * HH;         // [1024]

    // 1) zero accumulators
    const int nws = BB * PP * HH + BB * PP;              // 525,312 floats
    pp_zero_ws<<<(nws + 255) / 256, 256, 0, stream>>>(sums, nws);

    // 2) patch pooling (sorted-run accumulation + boundary atomics)
    pp_pool<<<BB * (SS / CS), PT, 0, stream>>>(byte_hiddens, patch_ids,
                                               sums, counts);

    // 3) WMMA projection: 3072 tiles, 8 waves/block -> 384 blocks
    pp_gemm_wmma<<<384, 256, 0, stream>>>(sums, counts, W_proj, b_proj, out);
}